// NN_57234734186711
// MI455X (gfx1250) — compile-verified
//
#include <hip/hip_runtime.h>
#include <hip/hip_bf16.h>

#define B_   128
#define IN_  512
#define HID_ 2048

typedef __attribute__((ext_vector_type(2))) float v2f;
typedef __attribute__((ext_vector_type(4))) float v4f;
typedef __attribute__((ext_vector_type(8))) float v8f;

// ---------------------------------------------------------------------------
// Kernel 1: std1[o,i] = exp(0.5 * w1_logvar[o,i])   (2048*512 = 1M elements)
// ---------------------------------------------------------------------------
__global__ __launch_bounds__(256) void std_precompute_kernel(
    const float* __restrict__ logvar, float* __restrict__ stdout_, int n) {
  int i = blockIdx.x * 256 + threadIdx.x;
  if (i < n) stdout_[i] = __expf(0.5f * logvar[i]);
}

// ---------------------------------------------------------------------------
// Kernel 2: mean[b,o] = sum_k x[b,k] * w1_mu[o,k]  via V_WMMA_F32_16X16X4_F32.
// One wave per 16x16 output tile; K swept 4 at a time (full fp32 precision).
// A-frag layout (ISA 7.12.2, 32-bit A 16x4): lane<16 -> M=lane, K={0,1};
// lane>=16 -> M=lane-16, K={2,3}.  B-frag mirrors with N striped over lanes.
// C/D: VGPR r -> row M = r + (lane>=16)*8, col N = lane&15.
// ---------------------------------------------------------------------------
__global__ __launch_bounds__(128) void mean_gemm_wmma_kernel(
    const float* __restrict__ X,      // [B_, IN_]
    const float* __restrict__ W,      // [HID_, IN_]
    float* __restrict__ meanOut) {    // [B_, HID_]
  const int tile = blockIdx.x * 4 + (threadIdx.x >> 5);  // 1024 tiles
  const int bt   = tile & 7;    // 8 tiles over B
  const int ot   = tile >> 3;   // 128 tiles over HID
  const int lane = threadIdx.x & 31;
  const int half = lane >> 4;
  const int l15  = lane & 15;

  const float* arow = X + (bt * 16 + l15) * IN_ + half * 2;  // A: row M=l15
  const float* brow = W + (ot * 16 + l15) * IN_ + half * 2;  // B: col N=l15

  v8f acc = {};
#pragma unroll 8
  for (int k0 = 0; k0 < IN_; k0 += 4) {
    v2f a = *(const v2f*)(arow + k0);
    v2f b = *(const v2f*)(brow + k0);
    // 8 args: (neg_a, A, neg_b, B, c_mod, C, reuse_a, reuse_b)
    acc = __builtin_amdgcn_wmma_f32_16x16x4_f32(
        false, a, false, b, (short)0, acc, false, false);
  }

  float* out = meanOut + (bt * 16 + half * 8) * HID_ + ot * 16 + l15;
#pragma unroll
  for (int r = 0; r < 8; ++r) out[r * HID_] = acc[r];
}

// ---------------------------------------------------------------------------
// Kernel 3 (bandwidth-critical): streams eps_w1 (512 MB, used once -> NT loads),
// computes noise[b,o] = sum_i x[b,i]*std1[o,i]*eps[b,o,i], fuses bias + relu.
// One wave per (b, 8 consecutive o). x-frag (16 floats) held in registers and
// reused across the 8 outputs; std1 re-read from L2 (4 MB resident, eps is NT
// so it does not evict it). Wave32 shfl_xor reduction.
// ---------------------------------------------------------------------------
__global__ __launch_bounds__(256) void noise_bias_relu_kernel(
    const float* __restrict__ x,        // [B_, IN_]
    const float* __restrict__ eps_w1,   // [B_, HID_, IN_]
    const float* __restrict__ eps_b1,   // [B_, HID_]
    const float* __restrict__ b1_mu,    // [HID_]
    const float* __restrict__ b1_logvar,// [HID_]
    const float* __restrict__ std1,     // [HID_, IN_]
    const float* __restrict__ meanIn,   // [B_, HID_]
    float* __restrict__ hOut) {         // [B_, HID_]
  const int lane = threadIdx.x & 31;
  const int gw   = blockIdx.x * 8 + (threadIdx.x >> 5);  // 32768 waves
  const int b    = gw >> 8;          // 0..127
  const int o0   = (gw & 255) * 8;   // 256 groups of 8 outputs
  const int ibase = lane * 4;        // 16B-aligned, coalesced across wave

  v4f xf[4];
#pragma unroll
  for (int j = 0; j < 4; ++j)
    xf[j] = *(const v4f*)(x + b * IN_ + ibase + j * 128);

#pragma unroll
  for (int oo = 0; oo < 8; ++oo) {
    const int o = o0 + oo;
    const float* srow = std1 + o * IN_ + ibase;
    const float* erow = eps_w1 + ((size_t)(b * HID_ + o)) * IN_ + ibase;
    float acc = 0.f;
#pragma unroll
    for (int j = 0; j < 4; ++j) {
      v4f s = *(const v4f*)(srow + j * 128);
      v4f e = __builtin_nontemporal_load((const v4f*)(erow + j * 128));
      v4f xs = xf[j] * s;
      acc += xs.x * e.x;
      acc += xs.y * e.y;
      acc += xs.z * e.z;
      acc += xs.w * e.w;
    }
#pragma unroll
    for (int m = 16; m > 0; m >>= 1) acc += __shfl_xor(acc, m, 32);
    if (lane == 0) {
      float bias = b1_mu[o] + __expf(0.5f * b1_logvar[o]) * eps_b1[b * HID_ + o];
      float v = meanIn[b * HID_ + o] + acc + bias;
      hOut[b * HID_ + o] = fmaxf(v, 0.f);
    }
  }
}

// ---------------------------------------------------------------------------
// Kernel 4: preds[b] = sum_o h[b,o]*(w2_mu[o] + exp(.5*w2_lv[o])*eps_w2[b,o])
//                      + b2_mu + exp(.5*b2_lv)*eps_b2[b]
// One 256-thread block per sample; LDS tree reduction. Also zero-fills the
// second (zeros) output.
// ---------------------------------------------------------------------------
__global__ __launch_bounds__(256) void layer2_kernel(
    const float* __restrict__ h,         // [B_, HID_]
    const float* __restrict__ w2_mu,     // [HID_]
    const float* __restrict__ w2_logvar, // [HID_]
    const float* __restrict__ eps_w2,    // [B_, HID_]
    const float* __restrict__ eps_b2,    // [B_]
    const float* __restrict__ b2_mu,     // [1]
    const float* __restrict__ b2_logvar, // [1]
    float* __restrict__ out) {           // [2*B_]
  __shared__ float red[256];
  const int b = blockIdx.x;
  const int t = threadIdx.x;
  float acc = 0.f;
#pragma unroll
  for (int o = t; o < HID_; o += 256) {
    float w = w2_mu[o] + __expf(0.5f * w2_logvar[o]) * eps_w2[b * HID_ + o];
    acc += h[b * HID_ + o] * w;
  }
  red[t] = acc;
  __syncthreads();
  for (int s = 128; s > 0; s >>= 1) {
    if (t < s) red[t] += red[t + s];
    __syncthreads();
  }
  if (t == 0) {
    float bias = b2_mu[0] + __expf(0.5f * b2_logvar[0]) * eps_b2[b];
    out[b] = red[0] + bias;
    out[B_ + b] = 0.f;  // second reference output is zeros_like(preds)
  }
}

// ---------------------------------------------------------------------------
extern "C" void kernel_launch(void* const* d_in, const int* in_sizes, int n_in,
                              void* d_out, int out_size, void* d_ws, size_t ws_size,
                              hipStream_t stream) {
  const float* x         = (const float*)d_in[0];
  const float* eps_w1    = (const float*)d_in[1];
  const float* eps_b1    = (const float*)d_in[2];
  const float* eps_w2    = (const float*)d_in[3];
  const float* eps_b2    = (const float*)d_in[4];
  const float* w1_mu     = (const float*)d_in[5];
  const float* w1_logvar = (const float*)d_in[6];
  const float* b1_mu     = (const float*)d_in[7];
  const float* b1_logvar = (const float*)d_in[8];
  const float* w2_mu     = (const float*)d_in[9];
  const float* w2_logvar = (const float*)d_in[10];
  const float* b2_mu     = (const float*)d_in[11];
  const float* b2_logvar = (const float*)d_in[12];
  float* out = (float*)d_out;

  char* ws = (char*)d_ws;
  float* std1  = (float*)(ws);                       // 4 MB: [HID_, IN_]
  float* meanB = (float*)(ws + (4u << 20));          // 1 MB: [B_, HID_]
  float* hB    = (float*)(ws + (5u << 20));          // 1 MB: [B_, HID_]

  // 1) std1 = exp(0.5*logvar): 1M elements
  std_precompute_kernel<<<(HID_ * IN_) / 256, 256, 0, stream>>>(
      w1_logvar, std1, HID_ * IN_);

  // 2) mean = x @ w1_mu^T via fp32 WMMA. 1024 16x16 tiles, 4 waves/block.
  mean_gemm_wmma_kernel<<<256, 128, 0, stream>>>(x, w1_mu, meanB);

  // 3) stream eps_w1 once (NT), fuse noise + bias + relu. 32768 waves.
  noise_bias_relu_kernel<<<4096, 256, 0, stream>>>(
      x, eps_w1, eps_b1, b1_mu, b1_logvar, std1, meanB, hB);

  // 4) final reduction per sample.
  layer2_kernel<<<B_, 256, 0, stream>>>(
      hB, w2_mu, w2_logvar, eps_w2, eps_b2, b2_mu, b2_logvar, out);
}